// RNNDecoder_57973468561893
// MI455X (gfx1250) — compile-verified
//
#include <hip/hip_runtime.h>
#include <hip/hip_bf16.h>
#include <math.h>

#define T_LEN   2048
#define DUNITS  1024
#define ATT_DIM 512
#define EPROJS  512
#define ODIM    32000
#define CHANS   10
#define FILTS   100
#define K_TAPS  (2 * FILTS + 1)
#define EMB     1024
#define SCALING 2.0f

typedef float v2f __attribute__((ext_vector_type(2)));
typedef float v8f __attribute__((ext_vector_type(8)));

// ---------------- wave32 reductions ----------------
__device__ static inline float wave_reduce_sum(float v) {
#pragma unroll
    for (int off = 16; off > 0; off >>= 1) v += __shfl_down(v, off, 32);
    return v;
}
__device__ static inline float wave_reduce_max(float v) {
#pragma unroll
    for (int off = 16; off > 0; off >>= 1) v = fmaxf(v, __shfl_down(v, off, 32));
    return v;
}

// =====================================================================
// WMMA fp32 GEMV:  y[j] = dot(W1[j,:], x1) (+ dot(W2[j,:], x2)) (+ b1[j] + b2[j])
// Each wave computes TILES 16-row tiles via chained V_WMMA_F32_16X16X4_F32.
// A-operand layout (ISA 16x4 f32): lane m<16 holds (W[r][k],W[r][k+1]),
// lane m+16 holds (W[r][k+2],W[r][k+3])  -> one global_load_b64 per lane.
// B-operand: x chunk replicated into all 16 columns (one ds_load_b64,
// broadcast), so D columns are replicas; lanes 0/16 each own 8 rows of D
// in statically indexed VGPRs. fp32 accumulate throughout.
// blockDim = 256 (8 waves). K1,K2 compile-time, multiples of 16.
// grid.x = M / (128*TILES); all Ms divide exactly.
// =====================================================================
template <int K1, int K2, int TILES>
__global__ void __launch_bounds__(256) gemv_wmma_kernel(
    const float* __restrict__ W1, const float* __restrict__ x1,
    const float* __restrict__ W2, const float* __restrict__ x2,
    const float* __restrict__ b1, const float* __restrict__ b2,
    float* __restrict__ y, int M)
{
    __shared__ float lds[K1 + K2];
    const int tid = threadIdx.x;

    for (int i = tid; i < K1; i += 256) lds[i] = x1[i];
    if constexpr (K2 > 0) {
        for (int i = tid; i < K2; i += 256) lds[K1 + i] = x2[i];
    }
    __syncthreads();

    const int wave = tid >> 5;
    const int lane = tid & 31;
    const int tileRow = (blockIdx.x * 8 + wave) * (16 * TILES);
    if (tileRow >= M) return;               // wave-uniform (EXEC all-1s at WMMA)

    const int rowInTile = lane & 15;
    const int kShift    = (lane >> 4) << 1; // 0 for lanes 0-15, 2 for lanes 16-31

    v8f acc[TILES];
#pragma unroll
    for (int t = 0; t < TILES; ++t) acc[t] = (v8f){0.f,0.f,0.f,0.f,0.f,0.f,0.f,0.f};

    {
        const float* base = W1 + (size_t)(tileRow + rowInTile) * K1 + kShift;
#pragma unroll 8
        for (int k = 0; k < K1; k += 4) {
            v2f b = *(const v2f*)(lds + k + kShift);
#pragma unroll
            for (int t = 0; t < TILES; ++t) {
                v2f a = *(const v2f*)(base + (size_t)(16 * t) * K1 + k);
                acc[t] = __builtin_amdgcn_wmma_f32_16x16x4_f32(
                    false, a, false, b, (short)0, acc[t], false, false);
            }
        }
    }
    if constexpr (K2 > 0) {
        const float* base = W2 + (size_t)(tileRow + rowInTile) * K2 + kShift;
#pragma unroll 8
        for (int k = 0; k < K2; k += 4) {
            v2f b = *(const v2f*)(lds + K1 + k + kShift);
#pragma unroll
            for (int t = 0; t < TILES; ++t) {
                v2f a = *(const v2f*)(base + (size_t)(16 * t) * K2 + k);
                acc[t] = __builtin_amdgcn_wmma_f32_16x16x4_f32(
                    false, a, false, b, (short)0, acc[t], false, false);
            }
        }
    }

    // D layout: lane 0 holds rows 0..7 (any column), lane 16 holds rows 8..15.
    if ((lane & 15) == 0) {
#pragma unroll
        for (int t = 0; t < TILES; ++t) {
            const int base = tileRow + 16 * t + ((lane >> 4) << 3);
#pragma unroll
            for (int v = 0; v < 8; ++v) {
                float val = acc[t][v];
                if (b1) val += b1[base + v];
                if (b2) val += b2[base + v];
                y[base + v] = val;
            }
        }
    }
}

// ---------------- embedding row gather into ey_cat[0:1024] ----------------
__global__ void __launch_bounds__(256) embed_kernel(
    const int* __restrict__ vy, const float* __restrict__ embW,
    float* __restrict__ eycat)
{
    int i = blockIdx.x * 256 + threadIdx.x;
    if (i < EMB) eycat[i] = embW[(size_t)vy[0] * EMB + i];
}

// ---------------- conv1d + attention energy e[t], one block per t ----------------
__global__ void __launch_bounds__(256) att_e_kernel(
    const float* __restrict__ a_prev, const float* __restrict__ conv_W,
    const float* __restrict__ att_W,  const float* __restrict__ pceh,
    const float* __restrict__ dec_z,  const float* __restrict__ gvec_W,
    const float* __restrict__ gvec_b, const float* __restrict__ mask,
    float* __restrict__ e_out)
{
    __shared__ float s_part[CHANS * 8];
    __shared__ float s_conv[CHANS];
    __shared__ float s_red[8];
    const int t    = blockIdx.x;
    const int tid  = threadIdx.x;
    const int wave = tid >> 5, lane = tid & 31;

    // cross-correlation taps (XLA conv = no kernel flip), zero padding FILTS
    float av = 0.f;
    if (tid < K_TAPS) {
        int pos = t + tid - FILTS;
        if (pos >= 0 && pos < T_LEN) av = a_prev[pos];
    }
#pragma unroll
    for (int c = 0; c < CHANS; ++c) {
        float v = (tid < K_TAPS) ? av * conv_W[c * K_TAPS + tid] : 0.f;
        v = wave_reduce_sum(v);
        if (lane == 0) s_part[c * 8 + wave] = v;
    }
    __syncthreads();
    if (tid < CHANS) {
        float s = 0.f;
#pragma unroll
        for (int w = 0; w < 8; ++w) s += s_part[tid * 8 + w];
        s_conv[tid] = s;
    }
    __syncthreads();

    float p = 0.f;
    for (int a = tid; a < ATT_DIM; a += 256) {
        float s = dec_z[a] + pceh[(size_t)t * ATT_DIM + a];
#pragma unroll
        for (int c = 0; c < CHANS; ++c) s += s_conv[c] * att_W[a * CHANS + c];
        p += tanhf(s) * gvec_W[a];
    }
    p = wave_reduce_sum(p);
    if (lane == 0) s_red[wave] = p;
    __syncthreads();
    if (tid == 0) {
        float s = 0.f;
#pragma unroll
        for (int w = 0; w < 8; ++w) s += s_red[w];
        e_out[t] = s + gvec_b[0] + mask[t];
    }
}

// ---------------- w = softmax(2*e) over T=2048, single block ----------------
__global__ void __launch_bounds__(1024) softmax_w_kernel(
    const float* __restrict__ e, float* __restrict__ w_ws, float* __restrict__ w_out)
{
    __shared__ float s_red[32];
    __shared__ float s_M, s_S;
    const int tid = threadIdx.x;
    const int wave = tid >> 5, lane = tid & 31;

    float z0 = SCALING * e[tid];
    float z1 = SCALING * e[tid + 1024];
    float m = fmaxf(z0, z1);
    m = wave_reduce_max(m);
    if (lane == 0) s_red[wave] = m;
    __syncthreads();
    if (wave == 0) {
        float v = s_red[lane];
        v = wave_reduce_max(v);
        if (lane == 0) s_M = v;
    }
    __syncthreads();
    const float M = s_M;
    float e0 = expf(z0 - M), e1 = expf(z1 - M);
    float s = e0 + e1;
    s = wave_reduce_sum(s);
    if (lane == 0) s_red[wave] = s;
    __syncthreads();
    if (wave == 0) {
        float v = s_red[lane];
        v = wave_reduce_sum(v);
        if (lane == 0) s_S = v;
    }
    __syncthreads();
    const float inv = 1.0f / s_S;
    float w0 = e0 * inv, w1 = e1 * inv;
    w_ws[tid] = w0;  w_ws[tid + 1024] = w1;
    w_out[tid] = w0; w_out[tid + 1024] = w1;
}

// ---------------- att_c partial sums over t-chunks (deterministic) ----------------
__global__ void __launch_bounds__(256) attc_part_kernel(
    const float* __restrict__ enc_h, const float* __restrict__ w,
    float* __restrict__ part) // [8][EPROJS]
{
    const int a  = blockIdx.x * 256 + threadIdx.x;  // 0..511
    const int t0 = blockIdx.y * 256;                // 8 chunks
    float acc = 0.f;
    for (int t = t0; t < t0 + 256; ++t)
        acc = fmaf(enc_h[(size_t)t * EPROJS + a], w[t], acc);
    part[blockIdx.y * EPROJS + a] = acc;
}

__global__ void __launch_bounds__(256) attc_reduce_kernel(
    const float* __restrict__ part, float* __restrict__ eycat)
{
    const int a = blockIdx.x * 256 + threadIdx.x;
    float s = 0.f;
#pragma unroll
    for (int c = 0; c < 8; ++c) s += part[c * EPROJS + a];
    eycat[EMB + a] = s;   // second half of [ey, att_c]
}

// ---------------- LSTM pointwise ----------------
__global__ void __launch_bounds__(256) lstm_act_kernel(
    const float* __restrict__ gates, const float* __restrict__ c_prev,
    float* __restrict__ c_out, float* __restrict__ h_out1, float* __restrict__ h_out2)
{
    const int j = blockIdx.x * 256 + threadIdx.x;
    if (j >= DUNITS) return;
    float ig = gates[j];
    float fg = gates[DUNITS + j];
    float gg = gates[2 * DUNITS + j];
    float og = gates[3 * DUNITS + j];
    float si = 1.f / (1.f + expf(-ig));
    float sf = 1.f / (1.f + expf(-fg));
    float so = 1.f / (1.f + expf(-og));
    float cn = sf * c_prev[j] + si * tanhf(gg);
    float hn = so * tanhf(cn);
    c_out[j]  = cn;
    h_out1[j] = hn;
    h_out2[j] = hn;
}

// ---------------- log_softmax over ODIM, single block ----------------
__global__ void __launch_bounds__(1024) logsoftmax_kernel(
    const float* __restrict__ logits, float* __restrict__ out)
{
    __shared__ float s_red[32];
    __shared__ float s_M, s_S;
    const int tid = threadIdx.x;
    const int wave = tid >> 5, lane = tid & 31;

    float m = -INFINITY;
    for (int i = tid; i < ODIM; i += 1024) m = fmaxf(m, logits[i]);
    m = wave_reduce_max(m);
    if (lane == 0) s_red[wave] = m;
    __syncthreads();
    if (wave == 0) {
        float v = s_red[lane];
        v = wave_reduce_max(v);
        if (lane == 0) s_M = v;
    }
    __syncthreads();
    const float M = s_M;
    float s = 0.f;
    for (int i = tid; i < ODIM; i += 1024) s += expf(logits[i] - M);
    s = wave_reduce_sum(s);
    if (lane == 0) s_red[wave] = s;
    __syncthreads();
    if (wave == 0) {
        float v = s_red[lane];
        v = wave_reduce_sum(v);
        if (lane == 0) s_S = v;
    }
    __syncthreads();
    const float lse = M + logf(s_S);
    for (int i = tid; i < ODIM; i += 1024) out[i] = logits[i] - lse;
}

// =====================================================================
extern "C" void kernel_launch(void* const* d_in, const int* in_sizes, int n_in,
                              void* d_out, int out_size, void* d_ws, size_t ws_size,
                              hipStream_t stream) {
    (void)in_sizes; (void)n_in; (void)out_size; (void)ws_size;

    const int*   vy      = (const int*)  d_in[0];
    const float* z_prev0 = (const float*)d_in[1];
    const float* z_prev1 = (const float*)d_in[2];
    const float* c_prev0 = (const float*)d_in[3];
    const float* c_prev1 = (const float*)d_in[4];
    const float* a_prev  = (const float*)d_in[5];
    const float* pceh    = (const float*)d_in[6];
    const float* enc_h   = (const float*)d_in[7];
    const float* mask    = (const float*)d_in[8];
    const float* embed_W = (const float*)d_in[9];
    const float* conv_W  = (const float*)d_in[10];
    const float* att_W   = (const float*)d_in[11];
    const float* dec_W   = (const float*)d_in[12];
    const float* gvec_W  = (const float*)d_in[13];
    const float* gvec_b  = (const float*)d_in[14];
    const float* W_ih0   = (const float*)d_in[15];
    const float* W_hh0   = (const float*)d_in[16];
    const float* b_ih0   = (const float*)d_in[17];
    const float* b_hh0   = (const float*)d_in[18];
    const float* W_ih1   = (const float*)d_in[19];
    const float* W_hh1   = (const float*)d_in[20];
    const float* b_ih1   = (const float*)d_in[21];
    const float* b_hh1   = (const float*)d_in[22];
    const float* out_W   = (const float*)d_in[23];
    const float* out_b   = (const float*)d_in[24];

    float* out = (float*)d_out;
    // output tuple layout: logp | c0 | c1 | z0 | z1 | w
    float* o_logp = out;
    float* o_c0   = out + ODIM;
    float* o_c1   = out + ODIM + DUNITS;
    float* o_z0   = out + ODIM + 2 * DUNITS;
    float* o_z1   = out + ODIM + 3 * DUNITS;
    float* o_w    = out + ODIM + 4 * DUNITS;

    // workspace layout (floats)
    float* ws       = (float*)d_ws;
    float* ws_eycat = ws;                 // 1536
    float* ws_decz  = ws + 1536;          // 512
    float* ws_e     = ws + 2048;          // 2048
    float* ws_w     = ws + 4096;          // 2048
    float* ws_part  = ws + 6144;          // 8*512
    float* ws_gates = ws + 10240;         // 4096
    float* ws_z0    = ws + 14336;         // 1024
    float* ws_z1    = ws + 15360;         // 1024
    float* ws_logit = ws + 16384;         // 32000

    // 1) ey = embed_W[vy] -> ey_cat[0:1024]
    embed_kernel<<<dim3(EMB / 256), dim3(256), 0, stream>>>(vy, embed_W, ws_eycat);

    // 2) dec_z = dec_W @ z_prev0   (WMMA GEMV: M=512, K=1024)
    gemv_wmma_kernel<DUNITS, 0, 1><<<dim3(512 / 128), dim3(256), 0, stream>>>(
        dec_W, z_prev0, nullptr, nullptr, nullptr, nullptr, ws_decz, 512);

    // 3) attention energies e[t]
    att_e_kernel<<<dim3(T_LEN), dim3(256), 0, stream>>>(
        a_prev, conv_W, att_W, pceh, ws_decz, gvec_W, gvec_b, mask, ws_e);

    // 4) w = softmax(2*e)
    softmax_w_kernel<<<dim3(1), dim3(1024), 0, stream>>>(ws_e, ws_w, o_w);

    // 5) att_c = w^T @ enc_h  (partials + reduce, deterministic)
    attc_part_kernel<<<dim3(EPROJS / 256, 8), dim3(256), 0, stream>>>(enc_h, ws_w, ws_part);
    attc_reduce_kernel<<<dim3(EPROJS / 256), dim3(256), 0, stream>>>(ws_part, ws_eycat);

    // 6) LSTM0 gates: W_ih0 @ [ey,att_c] + W_hh0 @ z_prev0 + biases (fused WMMA GEMV)
    gemv_wmma_kernel<EMB + EPROJS, DUNITS, 1><<<dim3(4096 / 128), dim3(256), 0, stream>>>(
        W_ih0, ws_eycat, W_hh0, z_prev0, b_ih0, b_hh0, ws_gates, 4096);
    lstm_act_kernel<<<dim3(DUNITS / 256), dim3(256), 0, stream>>>(
        ws_gates, c_prev0, o_c0, o_z0, ws_z0);

    // 7) LSTM1
    gemv_wmma_kernel<DUNITS, DUNITS, 1><<<dim3(4096 / 128), dim3(256), 0, stream>>>(
        W_ih1, ws_z0, W_hh1, z_prev1, b_ih1, b_hh1, ws_gates, 4096);
    lstm_act_kernel<<<dim3(DUNITS / 256), dim3(256), 0, stream>>>(
        ws_gates, c_prev1, o_c1, o_z1, ws_z1);

    // 8) logits = out_W @ z1 + out_b  (dominant 131MB stream: 2 tiles/wave, shared B)
    gemv_wmma_kernel<DUNITS, 0, 2><<<dim3(ODIM / 256), dim3(256), 0, stream>>>(
        out_W, ws_z1, nullptr, nullptr, out_b, nullptr, ws_logit, ODIM);

    // 9) logp = log_softmax(logits)
    logsoftmax_kernel<<<dim3(1), dim3(1024), 0, stream>>>(ws_logit, o_logp);
}